// NAFBlock_modified_15169824490026
// MI455X (gfx1250) — compile-verified
//
#include <hip/hip_runtime.h>
#include <math.h>

typedef float v2f __attribute__((ext_vector_type(2)));
typedef float v8f __attribute__((ext_vector_type(8)));

__device__ __forceinline__ float gelu_f(float x) {
    return 0.5f * x * (1.0f + tanhf(0.7978845608028654f * (x + 0.044715f * x * x * x)));
}

__device__ __forceinline__ v8f wmma_f32_4(v2f a, v2f b, v8f c) {
    // D = A(16x4,f32) x B(4x16,f32) + C(16x16,f32), exact fp32 matrix path
    return __builtin_amdgcn_wmma_f32_16x16x4_f32(false, a, false, b, (short)0, c,
                                                 false, false);
}

// Async DMA: 16 bytes global -> LDS per lane, tracked by ASYNCcnt (no VGPR data).
__device__ __forceinline__ void async_ld16(float* lds, const float* g) {
    uint32_t l = (uint32_t)(uintptr_t)lds;           // low 32 bits = LDS offset
    uint64_t a = (uint64_t)(uintptr_t)g;
    asm volatile("global_load_async_to_lds_b128 %0, %1, off"
                 :: "v"(l), "v"(a) : "memory");
}
__device__ __forceinline__ void async_wait0() {
    asm volatile("s_wait_asynccnt 0" ::: "memory");
}

// ---------------------------------------------------------------------------
// LayerNorm(channel dim, K=64) fused with 1x1 conv 64 -> 128 (WMMA GEMM)
// in:  [64][HW], W: [128][64] row-major, out: [128][HW]
// Per block: 64 pixels. 8 waves, each owns a 16-out-channel N-tile, 4 M-tiles.
// X tile staged via async global->LDS DMA (16B-aligned rows, stride 64).
// ---------------------------------------------------------------------------
__global__ __launch_bounds__(256) void ln_gemm64_128(
    const float* __restrict__ in, const float* __restrict__ lnw,
    const float* __restrict__ lnb, const float* __restrict__ Wm,
    const float* __restrict__ bias, float* __restrict__ out, int HW)
{
    __shared__ float smem[12352];
    float* sX = smem;           // 64 x 64 (16B-aligned rows for async DMA)
    float* sW = smem + 4096;    // 64 x 129 (B = W^T, padded)
    float* sC = smem;           // 128 x 66, aliases sX/sW after GEMM loop

    const int t  = threadIdx.x;
    const int p0 = blockIdx.x * 64;

    // 1) launch async DMA for the X tile (1024 16B chunks)
    for (int idx = t; idx < 1024; idx += 256) {
        int k = idx >> 4, c4 = (idx & 15) << 2;
        async_ld16(sX + k * 64 + c4, in + k * HW + p0 + c4);
    }
    // 2) stage W^T through VGPRs while the DMA is in flight
    for (int idx = t; idx < 128 * 64; idx += 256) {
        int o = idx >> 6, i = idx & 63;
        sW[i * 129 + o] = Wm[idx];
    }
    async_wait0();
    __syncthreads();

    if (t < 64) {                                        // LN per pixel
        float s = 0.f;
        for (int k = 0; k < 64; ++k) s += sX[k * 64 + t];
        float m = s * (1.0f / 64.0f);
        float v = 0.f;
        for (int k = 0; k < 64; ++k) { float d = sX[k * 64 + t] - m; v += d * d; }
        float r = rsqrtf(v * (1.0f / 64.0f) + 1e-6f);
        for (int k = 0; k < 64; ++k)
            sX[k * 64 + t] = (sX[k * 64 + t] - m) * r * lnw[k] + lnb[k];
    }
    __syncthreads();

    const int lane = t & 31, w = t >> 5, half = lane >> 4, l = lane & 15;
    const int n0 = w * 16;
    const v8f vz = {0.f, 0.f, 0.f, 0.f, 0.f, 0.f, 0.f, 0.f};
    v8f acc[4];
#pragma unroll
    for (int mt = 0; mt < 4; ++mt) acc[mt] = vz;

#pragma unroll
    for (int k0 = 0; k0 < 64; k0 += 4) {
        int ka = k0 + 2 * half;                          // ISA 16x4 A layout
        v2f b; b.x = sW[ka * 129 + n0 + l]; b.y = sW[(ka + 1) * 129 + n0 + l];
#pragma unroll
        for (int mt = 0; mt < 4; ++mt) {
            int m = mt * 16 + l;
            v2f a; a.x = sX[ka * 64 + m]; a.y = sX[(ka + 1) * 64 + m];
            acc[mt] = wmma_f32_4(a, b, acc[mt]);
        }
    }
    __syncthreads();                                     // sX/sW now dead
#pragma unroll
    for (int mt = 0; mt < 4; ++mt)
#pragma unroll
        for (int r = 0; r < 8; ++r)                      // documented C layout
            sC[(n0 + l) * 66 + mt * 16 + half * 8 + r] = acc[mt][r];
    __syncthreads();

    for (int r0 = 0; r0 < 128; r0 += 4) {                // coalesced store
        int r = r0 + (t >> 6), c = t & 63;
        out[r * HW + p0 + c] = sC[r * 66 + c] + bias[r];
    }
}

// ---------------------------------------------------------------------------
// x + basis_conv(de(en(x))) collapsed to Wc (128x128, identity folded in),
// fused with SimpleGate. X:[128][HW], Wc row-major [o][k], g:[64][HW]
// B fragments streamed from L2 (Wc is 64KB, L2-hot); A via async DMA to LDS.
// ---------------------------------------------------------------------------
__global__ __launch_bounds__(256) void qkv_gate128(
    const float* __restrict__ X, const float* __restrict__ Wc,
    float* __restrict__ g, int HW)
{
    __shared__ float smem[8448];
    float* sX = smem;            // 128 x 64 (16B-aligned rows for async DMA)
    float* sC = smem;            // 128 x 66 (aliases sX after GEMM loop)

    const int t  = threadIdx.x;
    const int p0 = blockIdx.x * 64;

    for (int idx = t; idx < 2048; idx += 256) {          // 128 rows x 16 chunks
        int k = idx >> 4, c4 = (idx & 15) << 2;
        async_ld16(sX + k * 64 + c4, X + k * HW + p0 + c4);
    }
    async_wait0();
    __syncthreads();

    const int lane = t & 31, w = t >> 5, half = lane >> 4, l = lane & 15;
    const int n0 = w * 16;
    const float* wrow = Wc + (n0 + l) * 128;             // B[k][n] = Wc[n][k]
    const v8f vz = {0.f, 0.f, 0.f, 0.f, 0.f, 0.f, 0.f, 0.f};
    v8f acc[4];
#pragma unroll
    for (int mt = 0; mt < 4; ++mt) acc[mt] = vz;

#pragma unroll 8
    for (int k0 = 0; k0 < 128; k0 += 4) {
        int ka = k0 + 2 * half;
        v2f b; b.x = wrow[ka]; b.y = wrow[ka + 1];
#pragma unroll
        for (int mt = 0; mt < 4; ++mt) {
            v2f a; a.x = sX[ka * 64 + mt * 16 + l];
            a.y = sX[(ka + 1) * 64 + mt * 16 + l];
            acc[mt] = wmma_f32_4(a, b, acc[mt]);
        }
    }
    __syncthreads();
#pragma unroll
    for (int mt = 0; mt < 4; ++mt)
#pragma unroll
        for (int r = 0; r < 8; ++r)
            sC[(n0 + l) * 66 + mt * 16 + half * 8 + r] = acc[mt][r];
    __syncthreads();

    for (int r0 = 0; r0 < 64; r0 += 4) {                 // SimpleGate epilogue
        int r = r0 + (t >> 6), c = t & 63;
        g[r * HW + p0 + c] = sC[r * 66 + c] * sC[(r + 64) * 66 + c];
    }
}

// ---------------------------------------------------------------------------
// out = res + rmul[o]*( (W·diag(svec)) @ X + bias[o] )   (conv3 / conv5)
// svec folded into the B operand so X can be a raw async DMA copy.
// ---------------------------------------------------------------------------
__global__ __launch_bounds__(256) void gemm_res64(
    const float* __restrict__ X, const float* __restrict__ svec,
    const float* __restrict__ Wm, const float* __restrict__ bias,
    const float* __restrict__ res, const float* __restrict__ rmul,
    float* __restrict__ out, int HW)
{
    __shared__ float smem[8256];
    float* sX = smem;            // 64 x 64 (16B-aligned rows for async DMA)
    float* sW = smem + 4096;     // 64 x 65
    float* sC = smem;            // 64 x 66 (aliases after GEMM loop)

    const int t  = threadIdx.x;
    const int p0 = blockIdx.x * 64;

    for (int idx = t; idx < 1024; idx += 256) {
        int k = idx >> 4, c4 = (idx & 15) << 2;
        async_ld16(sX + k * 64 + c4, X + k * HW + p0 + c4);
    }
    for (int idx = t; idx < 64 * 64; idx += 256) {       // B[k][n] = W[n][k]*s[k]
        int o = idx >> 6, i = idx & 63;
        float sc = svec ? svec[i] : 1.0f;
        sW[i * 65 + o] = Wm[idx] * sc;
    }
    async_wait0();
    __syncthreads();

    const int lane = t & 31, w = t >> 5, half = lane >> 4, l = lane & 15;
    const int n0 = (w & 3) * 16;
    const int mbase = (w >> 2) * 2;
    const v8f vz = {0.f, 0.f, 0.f, 0.f, 0.f, 0.f, 0.f, 0.f};
    v8f acc[2];
    acc[0] = vz; acc[1] = vz;

#pragma unroll
    for (int k0 = 0; k0 < 64; k0 += 4) {
        int ka = k0 + 2 * half;
        v2f b; b.x = sW[ka * 65 + n0 + l]; b.y = sW[(ka + 1) * 65 + n0 + l];
#pragma unroll
        for (int mi = 0; mi < 2; ++mi) {
            int m = (mbase + mi) * 16 + l;
            v2f a; a.x = sX[ka * 64 + m]; a.y = sX[(ka + 1) * 64 + m];
            acc[mi] = wmma_f32_4(a, b, acc[mi]);
        }
    }
    __syncthreads();
#pragma unroll
    for (int mi = 0; mi < 2; ++mi)
#pragma unroll
        for (int r = 0; r < 8; ++r)
            sC[(n0 + l) * 66 + (mbase + mi) * 16 + half * 8 + r] = acc[mi][r];
    __syncthreads();

    for (int r0 = 0; r0 < 64; r0 += 4) {
        int r = r0 + (t >> 6), c = t & 63;
        out[r * HW + p0 + c] =
            res[r * HW + p0 + c] + rmul[r] * (sC[r * 66 + c] + bias[r]);
    }
}

// ---------------------------------------------------------------------------
// Depthwise 3x3, pad 1.  in/out: [C][H][W], w9: [C][9]
// ---------------------------------------------------------------------------
__global__ __launch_bounds__(256) void dw3x3(
    const float* __restrict__ in, const float* __restrict__ w9,
    const float* __restrict__ b, float* __restrict__ out,
    int C, int H, int Wd)
{
    int n = blockIdx.x * 256 + threadIdx.x;
    int HW = H * Wd;
    if (n >= C * HW) return;
    int c = n / HW, p = n - c * HW;
    int h = p / Wd, x = p - h * Wd;
    const float* base = in + c * HW;
    const float* wk = w9 + c * 9;
    float a = b[c];
#pragma unroll
    for (int dy = -1; dy <= 1; ++dy) {
        int hh = h + dy;
        if (hh < 0 || hh >= H) continue;
#pragma unroll
        for (int dx = -1; dx <= 1; ++dx) {
            int ww = x + dx;
            if (ww < 0 || ww >= Wd) continue;
            a += wk[(dy + 1) * 3 + (dx + 1)] * base[hh * Wd + ww];
        }
    }
    out[n] = a;
}

// Per-channel mean over HW. One block per channel.
__global__ __launch_bounds__(256) void colmean(
    const float* __restrict__ in, float* __restrict__ out, int HW)
{
    __shared__ float red[256];
    int c = blockIdx.x, t = threadIdx.x;
    const float* row = in + (size_t)c * HW;
    float s = 0.f;
    for (int i = t; i < HW; i += 256) s += row[i];
    red[t] = s;
    __syncthreads();
    for (int off = 128; off > 0; off >>= 1) {
        if (t < off) red[t] += red[t + off];
        __syncthreads();
    }
    if (t == 0) out[c] = red[0] / (float)HW;
}

// y[o] = W[o,:]·x + bias[o]
__global__ void matvec(const float* __restrict__ Wm, const float* __restrict__ x,
                       const float* __restrict__ bias, float* __restrict__ y,
                       int O, int I)
{
    int o = threadIdx.x;
    if (o < O) {
        float a = bias ? bias[o] : 0.f;
        for (int i = 0; i < I; ++i) a += Wm[o * I + i] * x[i];
        y[o] = a;
    }
}

// Dynamic basis_conv weight generation: softmax attention -> Householder ->
// 2-layer MLP per (cout*cin) row.  wout is (cout,cin) row-major.
__global__ __launch_bounds__(256) void basis_weights(
    const float* __restrict__ s, const float* __restrict__ emb,
    const float* __restrict__ f1w, const float* __restrict__ f1b,
    const float* __restrict__ f2w, const float* __restrict__ f2b,
    const float* __restrict__ m1w, const float* __restrict__ m1b,
    const float* __restrict__ m2w, const float* __restrict__ m2b,
    float* __restrict__ wout, int cin, int M)
{
    __shared__ float hsh[32], ksh[32], red2[2];
    int t = threadIdx.x;
    if (t < 32) {
        float a = f1b[t];
        for (int i = 0; i < cin; ++i) a += f1w[t * cin + i] * s[i];
        hsh[t] = gelu_f(a);
    }
    __syncthreads();
    if (t < 32) {
        float a = f2b[t];
#pragma unroll
        for (int j = 0; j < 32; ++j) a += f2w[t * 32 + j] * hsh[j];
        ksh[t] = a;
    }
    __syncthreads();
    if (t == 0) {
        float mx = ksh[0];
        for (int j = 1; j < 32; ++j) mx = fmaxf(mx, ksh[j]);
        float sm = 0.f;
        for (int j = 0; j < 32; ++j) sm += expf(ksh[j] - mx);
        red2[0] = mx; red2[1] = sm;
    }
    __syncthreads();
    if (t < 32) ksh[t] = expf(ksh[t] - red2[0]) / red2[1];
    __syncthreads();

    int m = blockIdx.x * 256 + t;
    if (m < M) {
        float e[32]; float nn = 0.f;
#pragma unroll
        for (int j = 0; j < 32; ++j) { e[j] = emb[m * 32 + j]; nn += e[j] * e[j]; }
        float inv = rsqrtf(nn);
        float kp = 0.f;
#pragma unroll
        for (int j = 0; j < 32; ++j) kp += ksh[j] * e[j] * inv;
        float o = m2b[0];
#pragma unroll
        for (int f = 0; f < 16; ++f) {
            float a = m1b[f];
#pragma unroll
            for (int j = 0; j < 32; ++j)
                a += m1w[f * 32 + j] * (ksh[j] - 2.f * kp * e[j] * inv);
            o += m2w[f] * gelu_f(a);
        }
        wout[m] = o;
    }
}

// Wc[o][i] = sum_m Wde[o][m]*Wen[m][i] (+ identity to fold the x+qkv skip)
__global__ void combine_w(const float* __restrict__ Wde,
                          const float* __restrict__ Wen,
                          float* __restrict__ Wc,
                          int cout, int mid, int cin, int addI)
{
    int idx = blockIdx.x * 256 + threadIdx.x;
    if (idx < cout * cin) {
        int o = idx / cin, i = idx - o * cin;
        float a = 0.f;
        for (int m = 0; m < mid; ++m) a += Wde[o * mid + m] * Wen[m * cin + i];
        if (addI && o == i) a += 1.0f;
        Wc[idx] = a;
    }
}

// ---------------------------------------------------------------------------
extern "C" void kernel_launch(void* const* d_in, const int* in_sizes, int n_in,
                              void* d_out, int out_size, void* d_ws, size_t ws_size,
                              hipStream_t stream)
{
    (void)in_sizes; (void)n_in; (void)out_size; (void)ws_size;
    const int H = 384, Wd = 384, HW = H * Wd;

    const float* inp   = (const float*)d_in[0];
    const float* n1w   = (const float*)d_in[1];
    const float* n1b   = (const float*)d_in[2];
    const float* n2w   = (const float*)d_in[3];
    const float* n2b   = (const float*)d_in[4];
    const float* c1w   = (const float*)d_in[5];
    const float* c1b   = (const float*)d_in[6];
    const float* c2w   = (const float*)d_in[7];
    const float* c2b   = (const float*)d_in[8];
    const float* c3w   = (const float*)d_in[9];
    const float* c3b   = (const float*)d_in[10];
    const float* scaw  = (const float*)d_in[11];
    const float* scab  = (const float*)d_in[12];
    const float* c4w   = (const float*)d_in[13];
    const float* c4b   = (const float*)d_in[14];
    const float* c5w   = (const float*)d_in[15];
    const float* c5b   = (const float*)d_in[16];
    const float* beta  = (const float*)d_in[17];
    const float* gamma = (const float*)d_in[18];
    // bc param blocks: en1 @19, de1 @28, en2 @37, de2 @46 (9 arrays each)

    float* ws    = (float*)d_ws;
    float* bufA  = ws;                         // 128*HW : x1, later z
    float* bufB  = ws + (size_t)128 * HW;      // 128*HW : x2
    float* bufC  = ws + (size_t)256 * HW;      // 64*HW  : gate outputs
    float* bufY  = ws + (size_t)320 * HW;      // 64*HW  : y
    float* sm    = ws + (size_t)384 * HW;
    float* s0    = sm;                         // 128
    float* s1    = sm + 128;                   // 32
    float* Wen   = sm + 256;                   // 32x128
    float* Wde   = Wen + 4096;                 // 128x32
    float* Wc    = Wde + 4096;                 // 128x128
    float* scl   = Wc + 16384;                 // 64
    float* gmean = scl + 64;                   // 64

    const dim3 blk(256);
    const int gB = HW / 64;                    // 2304 GEMM blocks

#define BC(i) (const float*)d_in[(i)]
    // ---- first half-block ----
    ln_gemm64_128<<<gB, blk, 0, stream>>>(inp, n1w, n1b, c1w, c1b, bufA, HW);
    dw3x3<<<(128 * HW) / 256, blk, 0, stream>>>(bufA, c2w, c2b, bufB, 128, H, Wd);
    colmean<<<128, blk, 0, stream>>>(bufB, s0, HW);
    basis_weights<<<16, blk, 0, stream>>>(s0, BC(19), BC(20), BC(21), BC(22),
                                          BC(23), BC(24), BC(25), BC(26), BC(27),
                                          Wen, 128, 4096);
    matvec<<<1, blk, 0, stream>>>(Wen, s0, nullptr, s1, 32, 128);
    basis_weights<<<16, blk, 0, stream>>>(s1, BC(28), BC(29), BC(30), BC(31),
                                          BC(32), BC(33), BC(34), BC(35), BC(36),
                                          Wde, 32, 4096);
    combine_w<<<64, blk, 0, stream>>>(Wde, Wen, Wc, 128, 32, 128, 1);
    qkv_gate128<<<gB, blk, 0, stream>>>(bufB, Wc, bufC, HW);
    colmean<<<64, blk, 0, stream>>>(bufC, gmean, HW);
    matvec<<<1, blk, 0, stream>>>(scaw, gmean, scab, scl, 64, 64);
    gemm_res64<<<gB, blk, 0, stream>>>(bufC, scl, c3w, c3b, inp, beta, bufY, HW);

    // ---- second half-block ----
    ln_gemm64_128<<<gB, blk, 0, stream>>>(bufY, n2w, n2b, c4w, c4b, bufA, HW);
    colmean<<<128, blk, 0, stream>>>(bufA, s0, HW);
    basis_weights<<<16, blk, 0, stream>>>(s0, BC(37), BC(38), BC(39), BC(40),
                                          BC(41), BC(42), BC(43), BC(44), BC(45),
                                          Wen, 128, 4096);
    matvec<<<1, blk, 0, stream>>>(Wen, s0, nullptr, s1, 32, 128);
    basis_weights<<<16, blk, 0, stream>>>(s1, BC(46), BC(47), BC(48), BC(49),
                                          BC(50), BC(51), BC(52), BC(53), BC(54),
                                          Wde, 32, 4096);
    combine_w<<<64, blk, 0, stream>>>(Wde, Wen, Wc, 128, 32, 128, 1);
    qkv_gate128<<<gB, blk, 0, stream>>>(bufA, Wc, bufC, HW);
    gemm_res64<<<gB, blk, 0, stream>>>(bufC, nullptr, c5w, c5b, bufY, gamma,
                                       (float*)d_out, HW);
#undef BC
}